// KnowledgeLayer_8194797601191
// MI455X (gfx1250) — compile-verified
//
#include <hip/hip_runtime.h>
#include <stdint.h>

// Segmented log-sum-exp over sorted ix_out (E=33.5M edges, 2M segments).
// Preferred path: edge-partitioned blocks (no binary search), TDM
// tensor_load_to_lds (or per-lane ASYNCcnt DMA) staging, LDS
// ds_max_u32/ds_add_f32 segmented reduction, 64-bit CAS log-sum-exp merge
// for boundary segments. Fallback (if d_ws too small): segment-partitioned.

#define TPB     256     // 8 wave32 waves
#define ECHUNK  4096    // edges per block (main path)
#define MAXSPAN 1024    // max segment span handled via LDS tables
#define SPB     128     // segments per block (fallback path)
#define STAGE   4096    // staged-edge capacity (fallback path)

// ---------- helpers ----------------------------------------------------------

// Monotone order-preserving float -> uint mapping (for integer atomicMax)
__device__ __forceinline__ unsigned mapf(float f) {
    unsigned b = __float_as_uint(f);
    return (b & 0x80000000u) ? ~b : (b | 0x80000000u);
}
__device__ __forceinline__ float unmapf(unsigned m) {
    unsigned b = (m & 0x80000000u) ? (m & 0x7FFFFFFFu) : ~m;
    return __uint_as_float(b);
}
// mapf(-inf) == 0x007FFFFFu

// CDNA5 async global->LDS copy (ISA 15.18.3 op 98), ASYNCcnt-tracked.
#define ASYNC_LDS_B128(ldsOff, gaddr) \
    asm volatile("global_load_async_to_lds_b128 %0, %1, off" :: "v"(ldsOff), "v"(gaddr) : "memory")
#define WAIT_ASYNC0() asm volatile("s_wait_asynccnt 0" ::: "memory")

// ---- Tensor Data Mover (CDNA5 TDM, ISA ch.8): bulk 1-D global->LDS DMA -----
typedef unsigned int u32x4 __attribute__((ext_vector_type(4)));
typedef int          i32x4 __attribute__((ext_vector_type(4)));
typedef int          i32x8 __attribute__((ext_vector_type(8)));

#if defined(__has_builtin)
#if __has_builtin(__builtin_amdgcn_tensor_load_to_lds) && __has_builtin(__builtin_amdgcn_s_wait_tensorcnt)
#define HAVE_TDM 1
#if __has_include(<hip/amd_detail/amd_gfx1250_TDM.h>)
// amdgpu-toolchain (clang-23 / therock): 6-arg builtin
#define TDM_ISSUE(g0, g1) \
    __builtin_amdgcn_tensor_load_to_lds((g0), (g1), (i32x4)0, (i32x4)0, (i32x8)0, 0)
#else
// ROCm 7.2 (clang-22): 5-arg builtin
#define TDM_ISSUE(g0, g1) \
    __builtin_amdgcn_tensor_load_to_lds((g0), (g1), (i32x4)0, (i32x4)0, 0)
#endif
#endif
#endif

#ifdef HAVE_TDM
// 1-D tile of `nelem` u32 from global -> LDS. D# layout per ISA ch.8.
__device__ __forceinline__ void tdm_load_u32_1d(unsigned ldsAddr, const void* gptr, int nelem) {
    const uint64_t ga = (uint64_t)(uintptr_t)gptr;
    u32x4 g0;
    g0[0] = 1u;                                      // count=1, user mode
    g0[1] = ldsAddr;                                 // lds_addr (bytes)
    g0[2] = (unsigned)(ga & 0xFFFFFFFFull);          // global_addr[31:0]
    g0[3] = (unsigned)(ga >> 32) | 0x80000000u;      // global_addr[56:32] | type=2
    i32x8 g1;
    const unsigned n = (unsigned)nelem;
    g1[0] = 0x00020000;                              // data_size=4B, mask=0, no flags
    g1[1] = (int)((n & 0xFFFFu) << 16);              // tensor_dim0[15:0]
    g1[2] = (int)(((n >> 16) & 0xFFFFu) | (1u << 16)); // tensor_dim0[31:16], tensor_dim1=1
    g1[3] = (int)((n & 0xFFFFu) << 16);              // tile_dim0 = nelem (<=4096)
    g1[4] = 0;                                       // tile_dim1=0 (1-D), tile_dim2=0
    g1[5] = (int)n;                                  // tensor_dim0_stride[31:0]
    g1[6] = 0;                                       // stride hi, tensor_dim1_stride lo
    g1[7] = 0;
    TDM_ISSUE(g0, g1);
}
#endif

__device__ __forceinline__ unsigned long long pack_ms(float m, float s) {
    return ((unsigned long long)__float_as_uint(s) << 32) | (unsigned long long)__float_as_uint(m);
}

// Log-sum-exp partial merge via 64-bit CAS: (m,s) <- (m,s) (+) (m2,s2)
__device__ void cas_merge(unsigned long long* p, float m2, float s2) {
    unsigned long long cur = *p;
    for (;;) {
        const float m1 = __uint_as_float((unsigned)(cur & 0xFFFFFFFFull));
        const float s1 = __uint_as_float((unsigned)(cur >> 32));
        const float m  = fmaxf(m1, m2);
        float s = 0.0f;
        if (s1 > 0.0f) s += s1 * __expf(m1 - m);
        if (s2 > 0.0f) s += s2 * __expf(m2 - m);
        const unsigned long long nv   = pack_ms(m, s);
        const unsigned long long prev = atomicCAS(p, cur, nv);
        if (prev == cur) break;
        cur = prev;
    }
}

// ---------- main path: edge-partitioned --------------------------------------

__global__ __launch_bounds__(TPB)
void init_pairs_kernel(unsigned long long* __restrict__ pairs, int n) {
    const int i = (int)(blockIdx.x * blockDim.x + threadIdx.x);
    if (i < n) pairs[i] = 0x00000000FF800000ull;   // m = -inf, s = 0
}

__global__ __launch_bounds__(TPB)
void selse_edges_kernel(const float* __restrict__ x,
                        const int*   __restrict__ ix_in,
                        const int*   __restrict__ ix_out,
                        unsigned long long* __restrict__ pairs,
                        int E)
{
    __shared__ __align__(16) int s_ixo[ECHUNK];
    __shared__ __align__(16) int s_ixi[ECHUNK];   // overlaid with gathered values
    __shared__ unsigned s_maxbits[MAXSPAN];
    __shared__ float    s_sum[MAXSPAN];

    const int tid = (int)threadIdx.x;
    const int eb  = (int)blockIdx.x * ECHUNK;
    const int cnt = min(ECHUNK, E - eb);
    if (cnt <= 0) return;

    // ---- stage the block's edge slice into LDS
#ifdef HAVE_TDM
    if (tid == 0) {                                // TDM ignores EXEC: issue from one wave
        tdm_load_u32_1d((unsigned)(uintptr_t)(void*)s_ixo, ix_out + eb, cnt);
        tdm_load_u32_1d((unsigned)(uintptr_t)(void*)s_ixi, ix_in  + eb, cnt);
        __builtin_amdgcn_s_wait_tensorcnt(0);
    }
#else
    {
        const int n16 = cnt >> 2;                       // full 16B chunks
        const uint64_t gO = (uint64_t)(uintptr_t)(const void*)(ix_out + eb);
        const uint64_t gI = (uint64_t)(uintptr_t)(const void*)(ix_in  + eb);
        const unsigned lO = (unsigned)(uintptr_t)(void*)s_ixo;
        const unsigned lI = (unsigned)(uintptr_t)(void*)s_ixi;
        for (int c = tid; c < n16; c += TPB) {
            const unsigned boff = 16u * (unsigned)c;
            ASYNC_LDS_B128(lO + boff, gO + (uint64_t)boff);
            ASYNC_LDS_B128(lI + boff, gI + (uint64_t)boff);
        }
        for (int j = (n16 << 2) + tid; j < cnt; j += TPB) {   // tail (last block)
            s_ixo[j] = ix_out[eb + j];
            s_ixi[j] = ix_in[eb + j];
        }
        WAIT_ASYNC0();
    }
#endif
    __syncthreads();

    const int first = s_ixo[0];
    const int last  = s_ixo[cnt - 1];
    const int span  = last - first + 1;
    float* s_val = (float*)s_ixi;
    const int kEnd = (cnt + 7) >> 3;

    if (span <= MAXSPAN) {
        for (int i = tid; i < span; i += TPB) { s_maxbits[i] = 0x007FFFFFu; s_sum[i] = 0.0f; }
        __syncthreads();

        // Pass 1: per-segment max (run-merged), cache gathered values in LDS
        for (int k = tid; k < kEnd; k += TPB) {
            const int jlo = k << 3, jhi = min(jlo + 8, cnt);
            int curseg = -1; unsigned curmax = 0u;
            for (int j = jlo; j < jhi; ++j) {
                const int seg = s_ixo[j];
                const float v = x[s_ixi[j]];          // L2-resident gather (x = 16 MB)
                s_val[j] = v;
                const unsigned mv = mapf(v);
                if (seg != curseg) {
                    if (curseg >= 0) atomicMax(&s_maxbits[curseg - first], curmax);
                    curseg = seg; curmax = mv;
                } else if (mv > curmax) curmax = mv;
            }
            if (curseg >= 0) atomicMax(&s_maxbits[curseg - first], curmax);
        }
        __syncthreads();

        // Pass 2: sum of exp(v - local_max) per segment (run-merged)
        for (int k = tid; k < kEnd; k += TPB) {
            const int jlo = k << 3, jhi = min(jlo + 8, cnt);
            int curseg = -1; float csum = 0.0f, cmax = 0.0f;
            for (int j = jlo; j < jhi; ++j) {
                const int seg = s_ixo[j];
                const float v = s_val[j];
                if (seg != curseg) {
                    if (curseg >= 0) atomicAdd(&s_sum[curseg - first], csum);
                    curseg = seg;
                    cmax = unmapf(s_maxbits[seg - first]);
                    csum = __expf(v - cmax);
                } else {
                    csum += __expf(v - cmax);
                }
            }
            if (curseg >= 0) atomicAdd(&s_sum[curseg - first], csum);
        }
        __syncthreads();

        // Emit partials: interior segments owned solely by this block -> store;
        // the (<=2) boundary segments -> CAS log-sum-exp merge.
        for (int i = tid; i < span; i += TPB) {
            const float sv = s_sum[i];
            if (sv > 0.0f) {
                const float mv = unmapf(s_maxbits[i]);
                const int seg = first + i;
                if (seg == first || seg == last) cas_merge(&pairs[seg], mv, sv);
                else                             pairs[seg] = pack_ms(mv, sv);
            }
        }
    } else {
        // Rare fallback (span > MAXSPAN): per-run partials straight to CAS merge.
        for (int k = tid; k < kEnd; k += TPB) {
            const int jlo = k << 3, jhi = min(jlo + 8, cnt);
            int j = jlo;
            while (j < jhi) {
                const int seg = s_ixo[j];
                int r = j; float m = -__builtin_huge_valf();
                while (r < jhi && s_ixo[r] == seg) {
                    const float v = x[s_ixi[r]];
                    s_val[r] = v;
                    m = fmaxf(m, v);
                    ++r;
                }
                float ssum = 0.0f;
                for (int t = j; t < r; ++t) ssum += __expf(s_val[t] - m);
                cas_merge(&pairs[seg], m, ssum);
                j = r;
            }
        }
    }
}

__global__ __launch_bounds__(TPB)
void finalize_pairs_kernel(const unsigned long long* __restrict__ pairs,
                           float* __restrict__ out, int n) {
    const int i = (int)(blockIdx.x * blockDim.x + threadIdx.x);
    if (i < n) {
        const unsigned long long p = pairs[i];
        const float m = __uint_as_float((unsigned)(p & 0xFFFFFFFFull));
        const float s = __uint_as_float((unsigned)(p >> 32));
        out[i] = __logf(s + 1e-30f) + m;             // empty: log(eps)+(-inf) = -inf
    }
}

// ---------- fallback path: segment-partitioned (no workspace needed) ---------

__device__ __forceinline__ int lower_bound_i32(const int* __restrict__ a, int n, int key) {
    int lo = 0, hi = n;
    while (lo < hi) {
        const int mid = (lo + hi) >> 1;
        if (a[mid] < key) lo = mid + 1; else hi = mid;
    }
    return lo;
}

__global__ __launch_bounds__(TPB)
void seg_logsumexp_kernel(const float* __restrict__ x,
                          const int*   __restrict__ ix_in,
                          const int*   __restrict__ ix_out,
                          float*       __restrict__ out,
                          int E, int n_out)
{
    __shared__ unsigned s_maxbits[SPB];
    __shared__ float    s_sum[SPB];
    __shared__ __align__(16) int s_ixo[STAGE];
    __shared__ __align__(16) int s_ixi[STAGE];    // overlaid with values in pass 2
    __shared__ int s_e[2];

    const int tid     = (int)threadIdx.x;
    const int segbase = (int)blockIdx.x * SPB;
    const int segend  = min(segbase + SPB, n_out);
    const int nseg    = segend - segbase;

    if (tid < 2) s_e[tid] = lower_bound_i32(ix_out, E, (tid == 0) ? segbase : segend);
    if (tid < SPB) { s_maxbits[tid] = 0x007FFFFFu; s_sum[tid] = 0.0f; }
    __syncthreads();

    const int e0 = s_e[0], e1 = s_e[1];
    const int a0 = e0 & ~3, a1 = (e1 + 3) & ~3;
    const bool staged = (a1 - a0) <= STAGE;

    if (staged && e1 > e0) {
        const int n16 = (a1 - a0) >> 2;
        const uint64_t gO = (uint64_t)(uintptr_t)(const void*)(ix_out + a0);
        const uint64_t gI = (uint64_t)(uintptr_t)(const void*)(ix_in  + a0);
        const unsigned lO = (unsigned)(uintptr_t)(void*)s_ixo;
        const unsigned lI = (unsigned)(uintptr_t)(void*)s_ixi;
        for (int c = tid; c < n16; c += TPB) {
            const unsigned boff = 16u * (unsigned)c;
            ASYNC_LDS_B128(lO + boff, gO + (uint64_t)boff);
            ASYNC_LDS_B128(lI + boff, gI + (uint64_t)boff);
        }
        WAIT_ASYNC0();
    }
    __syncthreads();

    float* s_val = (float*)s_ixi;
    const int kBeg = e0 >> 3, kEnd = (e1 + 7) >> 3;

    for (int k = kBeg + tid; k < kEnd; k += TPB) {
        const int jlo = max(k << 3, e0), jhi = min((k << 3) + 8, e1);
        int curseg = -1; unsigned curmax = 0u;
        for (int j = jlo; j < jhi; ++j) {
            int seg, ii;
            if (staged) { seg = s_ixo[j - a0]; ii = s_ixi[j - a0]; }
            else        { seg = ix_out[j];     ii = ix_in[j];      }
            const float v = x[ii];
            if (staged) s_val[j - a0] = v;
            const unsigned mv = mapf(v);
            if (seg != curseg) {
                if (curseg >= 0) atomicMax(&s_maxbits[curseg - segbase], curmax);
                curseg = seg; curmax = mv;
            } else if (mv > curmax) curmax = mv;
        }
        if (curseg >= 0) atomicMax(&s_maxbits[curseg - segbase], curmax);
        if (!staged && (k + TPB) < kEnd) __builtin_prefetch(ix_out + ((k + TPB) << 3), 0, 0);
    }
    __syncthreads();

    for (int k = kBeg + tid; k < kEnd; k += TPB) {
        const int jlo = max(k << 3, e0), jhi = min((k << 3) + 8, e1);
        int curseg = -1; float csum = 0.0f, cmax = 0.0f;
        for (int j = jlo; j < jhi; ++j) {
            int seg; float v;
            if (staged) { seg = s_ixo[j - a0]; v = s_val[j - a0]; }
            else        { seg = ix_out[j];     v = x[ix_in[j]];   }
            if (seg != curseg) {
                if (curseg >= 0) atomicAdd(&s_sum[curseg - segbase], csum);
                curseg = seg;
                cmax = unmapf(s_maxbits[seg - segbase]);
                csum = __expf(v - cmax);
            } else csum += __expf(v - cmax);
        }
        if (curseg >= 0) atomicAdd(&s_sum[curseg - segbase], csum);
    }
    __syncthreads();

    if (tid < nseg) {
        const float m = unmapf(s_maxbits[tid]);
        out[segbase + tid] = __logf(s_sum[tid] + 1e-30f) + m;
    }
}

// ---------- dispatch ---------------------------------------------------------

extern "C" void kernel_launch(void* const* d_in, const int* in_sizes, int n_in,
                              void* d_out, int out_size, void* d_ws, size_t ws_size,
                              hipStream_t stream) {
    (void)n_in;
    const float* x      = (const float*)d_in[0];
    const int*   ix_in  = (const int*)d_in[1];
    const int*   ix_out = (const int*)d_in[2];
    float*       out    = (float*)d_out;
    const int E     = in_sizes[1];
    const int n_out = out_size;

    if (d_ws != nullptr && ws_size >= (size_t)n_out * sizeof(unsigned long long)) {
        unsigned long long* pairs = (unsigned long long*)d_ws;
        const int gridN = (n_out + TPB - 1) / TPB;
        const int gridE = (E + ECHUNK - 1) / ECHUNK;
        init_pairs_kernel<<<gridN, TPB, 0, stream>>>(pairs, n_out);
        selse_edges_kernel<<<gridE, TPB, 0, stream>>>(x, ix_in, ix_out, pairs, E);
        finalize_pairs_kernel<<<gridN, TPB, 0, stream>>>(pairs, out, n_out);
    } else {
        const int nblocks = (n_out + SPB - 1) / SPB;
        seg_logsumexp_kernel<<<nblocks, TPB, 0, stream>>>(x, ix_in, ix_out, out, E, n_out);
    }
}